// TAMH_9852654977291
// MI455X (gfx1250) — compile-verified
//
#include <hip/hip_runtime.h>
#include <math.h>

// ---------------- problem constants ----------------
#define BB   96
#define CC   128
#define HH   64
#define WW   128
#define HWSZ (HH*WW)          // 8192
#define TSZ  16
#define KF   1152             // C*K*K
#define KH   576              // KF/2
#define FUS  32
#define BN_EPS 1e-5f

// d_out layout: z (B*H*W) | H_corr (B*H*W) | R (B*C*16*16)
#define Z_OFF   0
#define HC_OFF  (BB*HWSZ)               // 786432
#define R_OFF   (2*BB*HWSZ)             // 1572864

typedef __attribute__((ext_vector_type(2))) float v2f;
typedef __attribute__((ext_vector_type(8))) float v8f;

// ================= K1: per-batch argmax of H_piece (first-index tie-break) ==========
__global__ void k_argmax(const float* __restrict__ hp, int* __restrict__ idx_out) {
  int b = blockIdx.x;
  const float* p = hp + (size_t)b * HWSZ;
  __shared__ float sv[256];
  __shared__ int   si[256];
  int t = threadIdx.x;
  float bv = -INFINITY; int bi = 0;
  for (int i = t; i < HWSZ; i += 256) {
    float v = p[i];
    if (v > bv) { bv = v; bi = i; }
  }
  sv[t] = bv; si[t] = bi;
  __syncthreads();
  for (int s = 128; s > 0; s >>= 1) {
    if (t < s) {
      float ov = sv[t + s]; int oi = si[t + s];
      if (ov > sv[t] || (ov == sv[t] && oi < si[t])) { sv[t] = ov; si[t] = oi; }
    }
    __syncthreads();
  }
  if (t == 0) idx_out[b] = si[0];
}

// ================= K2: bilinear template extract R + channel means (gap) ============
__global__ void k_template(const float* __restrict__ Hf, const int* __restrict__ idxs,
                           float* __restrict__ Rout, float* __restrict__ gap) {
  int bc = blockIdx.x;             // b*C + c
  int b  = bc >> 7;                // C == 128
  int t  = threadIdx.x;
  int idx = idxs[b];
  float u = (float)(idx & (WW - 1));
  float v = (float)(idx >> 7);
  int r = t >> 4, cidx = t & 15;
  float fy = v * (64.0f/63.0f)  - 0.5f + ((float)r    - 7.5f);
  float fx = u * (128.0f/127.0f) - 0.5f + ((float)cidx - 7.5f);
  fy = fminf(fmaxf(fy, 0.0f), 63.0f);
  fx = fminf(fmaxf(fx, 0.0f), 127.0f);
  float y0f = floorf(fy), x0f = floorf(fx);
  float wy = fy - y0f, wx = fx - x0f;
  int y0 = min(max((int)y0f, 0), HH - 1);
  int x0 = min(max((int)x0f, 0), WW - 1);
  int y1 = min(y0 + 1, HH - 1);
  int x1 = min(x0 + 1, WW - 1);
  const float* img = Hf + (size_t)bc * HWSZ;
  float v00 = img[y0*WW + x0], v01 = img[y0*WW + x1];
  float v10 = img[y1*WW + x0], v11 = img[y1*WW + x1];
  float val = v00*(1.0f-wx)*(1.0f-wy) + v01*wx*(1.0f-wy)
            + v10*(1.0f-wx)*wy        + v11*wx*wy;
  Rout[(size_t)bc * 256 + t] = val;
  __shared__ float ssum[256];
  ssum[t] = val;
  __syncthreads();
  for (int s = 128; s > 0; s >>= 1) {
    if (t < s) ssum[t] += ssum[t + s];
    __syncthreads();
  }
  if (t == 0) gap[bc] = ssum[0] * (1.0f/256.0f);
}

// ================= K3: fp32 WMMA GEMM  out(M,N) = act(A(M,Kd) @ Wt(N,Kd)^T + bias) ===
// one wave computes one 16x16 tile using V_WMMA_F32_16X16X4_F32
__global__ void k_gemm_wmma(const float* __restrict__ A, const float* __restrict__ Wt,
                            const float* __restrict__ bias, float* __restrict__ out,
                            int Kd, int N, int ntiles_n, int ntiles_total, int relu) {
  int wslot = threadIdx.x >> 5;
  int wid = blockIdx.x * (blockDim.x >> 5) + wslot;
  if (wid >= ntiles_total) return;            // wave-uniform
  int lane = threadIdx.x & 31;
  int lm = lane & 15, hi = lane >> 4;
  int n0 = (wid % ntiles_n) * 16;
  int m0 = (wid / ntiles_n) * 16;
  const float* arow = A  + (size_t)(m0 + lm) * Kd + 2*hi;
  const float* brow = Wt + (size_t)(n0 + lm) * Kd + 2*hi;
  v8f acc = {};
  for (int k = 0; k < Kd; k += 4) {
    v2f a, bm;
    a.x  = arow[k];  a.y  = arow[k + 1];
    bm.x = brow[k];  bm.y = brow[k + 1];
    acc = __builtin_amdgcn_wmma_f32_16x16x4_f32(false, a, false, bm,
                                                (short)0, acc, false, false);
  }
  float bb = bias[n0 + lm];
  #pragma unroll
  for (int j = 0; j < 8; ++j) {
    float vv = acc[j] + bb;
    if (relu) vv = fmaxf(vv, 0.0f);
    out[(size_t)(m0 + j + 8*hi) * N + (n0 + lm)] = vv;
  }
}

// ================= K4: per-(b,c) 3x3 kernel L2-normalization =========================
__global__ void k_norm(const float* __restrict__ kp, float* __restrict__ wd, int total) {
  int t = blockIdx.x * blockDim.x + threadIdx.x;
  if (t >= total) return;
  const float* p = kp + (size_t)t * 9;
  float s = 0.0f;
  #pragma unroll
  for (int j = 0; j < 9; ++j) s += p[j] * p[j];
  float inv = 1.0f / fmaxf(sqrtf(s), 1e-12f);
  float* q = wd + (size_t)t * 9;
  #pragma unroll
  for (int j = 0; j < 9; ++j) q[j] = p[j] * inv;
}

// ================= K5: per-batch dynamic 3x3 correlation (128ch -> 1) + sigmoid ======
// LDS-tiled band kernel: block = (batch b, 8-row band). Memory-bound streamer.
#define BAND 8
__global__ void k_corr(const float* __restrict__ Hf, const float* __restrict__ wd,
                       float* __restrict__ hcorr) {
  int b  = blockIdx.y;
  int y0 = blockIdx.x * BAND;
  int tid = threadIdx.x;
  __shared__ float wsm[KF];            // 1152 weights for this batch
  __shared__ float tile[10 * 130];     // (BAND+2) x (W+2) halo tile
  for (int i = tid; i < KF; i += 256) wsm[i] = wd[(size_t)b * KF + i];
  int x  = tid & (WW - 1);
  int yl = tid >> 7;                   // 0 or 1
  float acc[4] = {0.f, 0.f, 0.f, 0.f};
  const float* base = Hf + (size_t)b * CC * HWSZ;
  for (int c = 0; c < CC; ++c) {
    __syncthreads();                   // tile reuse guard (also covers wsm on c==0)
    const float* ch = base + (size_t)c * HWSZ;
    for (int i = tid; i < 1300; i += 256) {
      int rr = i / 130, ccc = i - rr * 130;
      int gy = y0 - 1 + rr, gx = ccc - 1;
      float v = 0.0f;
      if (gy >= 0 && gy < HH && gx >= 0 && gx < WW) v = ch[gy * WW + gx];
      tile[i] = v;
    }
    // prefetch next channel's band (global_prefetch_b8), 40 cachelines
    if (c + 1 < CC && tid < 40) {
      int rr = tid >> 2, seg = tid & 3;
      int gy = y0 - 1 + rr;
      if (gy >= 0 && gy < HH)
        __builtin_prefetch(ch + HWSZ + gy * WW + seg * 32, 0, 0);
    }
    __syncthreads();
    const float* wc = &wsm[c * 9];
    float w00 = wc[0], w01 = wc[1], w02 = wc[2];
    float w10 = wc[3], w11 = wc[4], w12 = wc[5];
    float w20 = wc[6], w21 = wc[7], w22 = wc[8];
    int lb = yl * 130 + x;
    float ra0 = tile[lb + 0],       ra1 = tile[lb + 1],       ra2 = tile[lb + 2];
    float rb0 = tile[lb + 130 + 0], rb1 = tile[lb + 130 + 1], rb2 = tile[lb + 130 + 2];
    #pragma unroll
    for (int rr = 0; rr < 4; ++rr) {
      int ro = lb + (2 * rr + 2) * 130;
      float rc0 = tile[ro + 0], rc1 = tile[ro + 1], rc2 = tile[ro + 2];
      acc[rr] += ra0*w00 + ra1*w01 + ra2*w02
               + rb0*w10 + rb1*w11 + rb2*w12
               + rc0*w20 + rc1*w21 + rc2*w22;
      if (rr < 3) {
        ra0 = rc0; ra1 = rc1; ra2 = rc2;
        int rn = lb + (2 * rr + 3) * 130;
        rb0 = tile[rn + 0]; rb1 = tile[rn + 1]; rb2 = tile[rn + 2];
      }
    }
  }
  #pragma unroll
  for (int rr = 0; rr < 4; ++rr) {
    int y = y0 + yl + 2 * rr;
    hcorr[(size_t)b * HWSZ + y * WW + x] = 1.0f / (1.0f + expf(-acc[rr]));
  }
}

// ================= K6: fusion head via WMMA (im2col K=18->20, 2 N-tiles of 16 ch) ====
__device__ __forceinline__ float patchval(const float* __restrict__ g,
                                          const float* __restrict__ hc,
                                          int y, int x, int k) {
  if (k >= 18) return 0.0f;
  int cin = k / 9, rr = k - cin * 9;
  int yy = y + rr / 3 - 1;
  int xx = x + rr % 3 - 1;
  if (yy < 0 || yy >= HH || xx < 0 || xx >= WW) return 0.0f;
  const float* src = cin ? hc : g;
  return src[yy * WW + xx];
}

__global__ void k_fusion(const float* __restrict__ hgap, const float* __restrict__ hcorr,
                         const float* __restrict__ w1, const float* __restrict__ b1,
                         const float* __restrict__ gamma, const float* __restrict__ beta,
                         const float* __restrict__ mean,  const float* __restrict__ var,
                         const float* __restrict__ w2, const float* __restrict__ b2,
                         float* __restrict__ zout, int nwaves_total) {
  int wslot = threadIdx.x >> 5;
  int wid = blockIdx.x * (blockDim.x >> 5) + wslot;
  if (wid >= nwaves_total) return;           // wave-uniform
  int b    = wid / (HH * (WW / 16));
  int remn = wid % (HH * (WW / 16));
  int y    = remn / (WW / 16);
  int x0   = (remn % (WW / 16)) * 16;
  int lane = threadIdx.x & 31;
  int lm = lane & 15, hi = lane >> 4;
  const float* g  = hgap  + (size_t)b * HWSZ;
  const float* hc = hcorr + (size_t)b * HWSZ;
  int px = x0 + lm;                          // this lane's A-matrix row (pixel)
  v8f c0 = {}, c1 = {};
  #pragma unroll
  for (int ks = 0; ks < 5; ++ks) {           // K = 20 (18 real + 2 zero pad)
    int kb = ks * 4 + 2 * hi;
    v2f a, bm0, bm1;
    a.x = patchval(g, hc, y, px, kb);
    a.y = patchval(g, hc, y, px, kb + 1);
    bm0.x = (kb     < 18) ? w1[lm * 18 + kb]            : 0.0f;
    bm0.y = (kb + 1 < 18) ? w1[lm * 18 + kb + 1]        : 0.0f;
    bm1.x = (kb     < 18) ? w1[(lm + 16) * 18 + kb]     : 0.0f;
    bm1.y = (kb + 1 < 18) ? w1[(lm + 16) * 18 + kb + 1] : 0.0f;
    c0 = __builtin_amdgcn_wmma_f32_16x16x4_f32(false, a, false, bm0,
                                               (short)0, c0, false, false);
    c1 = __builtin_amdgcn_wmma_f32_16x16x4_f32(false, a, false, bm1,
                                               (short)0, c1, false, false);
  }
  // epilogue: bias + BN + swish, then 1x1 conv as cross-lane (N-dim) reduction
  int o0 = lm, o1 = lm + 16;
  float s0 = gamma[o0] * rsqrtf(var[o0] + BN_EPS);
  float s1 = gamma[o1] * rsqrtf(var[o1] + BN_EPS);
  float m0v = mean[o0], m1v = mean[o1];
  float be0 = beta[o0], be1 = beta[o1];
  float bb0 = b1[o0],  bb1 = b1[o1];
  float w20 = w2[o0],  w21 = w2[o1];
  float zsum[8];
  #pragma unroll
  for (int j = 0; j < 8; ++j) {
    float y0v = (c0[j] + bb0 - m0v) * s0 + be0;
    float y1v = (c1[j] + bb1 - m1v) * s1 + be1;
    float sw0 = y0v / (1.0f + expf(-y0v));
    float sw1 = y1v / (1.0f + expf(-y1v));
    zsum[j] = sw0 * w20 + sw1 * w21;
  }
  #pragma unroll
  for (int j = 0; j < 8; ++j) {
    #pragma unroll
    for (int m = 1; m < 16; m <<= 1)
      zsum[j] += __shfl_xor(zsum[j], m, 32);
  }
  if (lm == 0) {                              // lane 0: pixels x0..x0+7, lane 16: x0+8..x0+15
    float bz = b2[0];
    #pragma unroll
    for (int j = 0; j < 8; ++j) {
      float zz = 1.0f / (1.0f + expf(-(zsum[j] + bz)));
      zout[(size_t)b * HWSZ + y * WW + x0 + 8 * hi + j] = zz;
    }
  }
}

// ================= host-side launch ==================================================
extern "C" void kernel_launch(void* const* d_in, const int* in_sizes, int n_in,
                              void* d_out, int out_size, void* d_ws, size_t ws_size,
                              hipStream_t stream) {
  const float* Hf      = (const float*)d_in[0];
  const float* H_piece = (const float*)d_in[1];
  const float* H_gap   = (const float*)d_in[2];
  const float* fc1_w   = (const float*)d_in[3];
  const float* fc1_b   = (const float*)d_in[4];
  const float* fc2_w   = (const float*)d_in[5];
  const float* fc2_b   = (const float*)d_in[6];
  const float* conv1_w = (const float*)d_in[7];
  const float* conv1_b = (const float*)d_in[8];
  const float* bn_g    = (const float*)d_in[9];
  const float* bn_b    = (const float*)d_in[10];
  const float* bn_m    = (const float*)d_in[11];
  const float* bn_v    = (const float*)d_in[12];
  const float* conv2_w = (const float*)d_in[13];
  const float* conv2_b = (const float*)d_in[14];

  float* out = (float*)d_out;
  float* zout  = out + Z_OFF;
  float* hcorr = out + HC_OFF;
  float* Rout  = out + R_OFF;

  // workspace layout (floats)
  int*   idxs = (int*)d_ws;
  float* wsf  = (float*)d_ws;
  float* gap = wsf + 128;                    // B*C        = 12288
  float* hdn = gap + BB * CC;                // B*KH       = 55296
  float* kp  = hdn + BB * KH;                // B*KF       = 110592
  float* wd  = kp  + BB * KF;                // B*KF       = 110592

  // 1) argmax per batch
  k_argmax<<<BB, 256, 0, stream>>>(H_piece, idxs);

  // 2) template extract + gap
  k_template<<<BB * CC, 256, 0, stream>>>(Hf, idxs, Rout, gap);

  // 3) FC1 (relu) then FC2, fp32 WMMA
  {
    int ntn = KH / 16, ntt = (BB / 16) * ntn;             // 36 x 6 = 216
    k_gemm_wmma<<<(ntt + 7) / 8, 256, 0, stream>>>(gap, fc1_w, fc1_b, hdn,
                                                   CC, KH, ntn, ntt, 1);
  }
  {
    int ntn = KF / 16, ntt = (BB / 16) * ntn;             // 72 x 6 = 432
    k_gemm_wmma<<<(ntt + 7) / 8, 256, 0, stream>>>(hdn, fc2_w, fc2_b, kp,
                                                   KH, KF, ntn, ntt, 0);
  }

  // 4) per-(b,c) kernel normalization
  k_norm<<<(BB * CC + 255) / 256, 256, 0, stream>>>(kp, wd, BB * CC);

  // 5) dynamic correlation conv + sigmoid (memory-bound streamer over Hf)
  k_corr<<<dim3(HH / BAND, BB), 256, 0, stream>>>(Hf, wd, hcorr);

  // 6) fusion head via WMMA
  {
    int nw = BB * HH * (WW / 16);                         // 49152 waves
    k_fusion<<<(nw + 7) / 8, 256, 0, stream>>>(H_gap, hcorr, conv1_w, conv1_b,
                                               bn_g, bn_b, bn_m, bn_v,
                                               conv2_w, conv2_b, zout, nw);
  }
}